// MaskToROI_8924942041139
// MI455X (gfx1250) — compile-verified
//
#include <hip/hip_runtime.h>
#include <hip/hip_bf16.h>
#include <stdint.h>

// Problem constants from the reference (mask: [16, 512, 512] fp32)
constexpr int B_DIM   = 16;
constexpr int H_DIM   = 512;
constexpr int W_DIM   = 512;
constexpr int N_ELEM  = B_DIM * H_DIM * W_DIM;   // 4,194,304
constexpr int HW      = H_DIM * W_DIM;           // 262,144 = 2^18
constexpr int ROI_HALF = 16;
constexpr float THRESH = 0.5f;

constexpr int THREADS = 256;                     // 8 waves (wave32)
constexpr int ITEMS   = 16;                      // elements per thread
constexpr int CHUNK   = THREADS * ITEMS;         // 4096 elements = 16 KB
constexpr int NBLK    = N_ELEM / CHUNK;          // 1024 chunks
constexpr int NWAVES  = THREADS / 32;            // 8 waves per block

// ---------------------------------------------------------------------------
// Kernel 1: per-chunk hit count
// ---------------------------------------------------------------------------
__global__ void count_kernel(const float* __restrict__ mask,
                             unsigned* __restrict__ counts) {
    __shared__ unsigned red[NWAVES];
    const int tid = threadIdx.x;
    const long base = (long)blockIdx.x * CHUNK;

    // Coalesced float4 reads: 16 B per lane per iteration.
    const float4* m4 = (const float4*)(mask + base);
    unsigned c = 0;
#pragma unroll
    for (int i = 0; i < ITEMS / 4; ++i) {
        float4 v = m4[i * THREADS + tid];
        c += (v.x > THRESH) + (v.y > THRESH) + (v.z > THRESH) + (v.w > THRESH);
    }
    // wave32 shuffle reduction
#pragma unroll
    for (int off = 16; off > 0; off >>= 1)
        c += __shfl_down(c, off, 32);
    if ((tid & 31) == 0) red[tid >> 5] = c;
    __syncthreads();
    if (tid == 0) {
        unsigned t = 0;
#pragma unroll
        for (int w = 0; w < NWAVES; ++w) t += red[w];
        counts[blockIdx.x] = t;
    }
}

// ---------------------------------------------------------------------------
// Kernel 2: exclusive scan over the NBLK block counts (single 1024-thread block)
// ---------------------------------------------------------------------------
__global__ void scan_kernel(const unsigned* __restrict__ counts,
                            unsigned* __restrict__ offsets,
                            unsigned* __restrict__ total) {
    __shared__ unsigned s[NBLK];
    const int t = threadIdx.x;
    const unsigned v = counts[t];
    s[t] = v;
    __syncthreads();
    for (int off = 1; off < NBLK; off <<= 1) {
        unsigned x = (t >= off) ? s[t - off] : 0u;
        __syncthreads();
        s[t] += x;
        __syncthreads();
    }
    offsets[t] = s[t] - v;                 // exclusive prefix
    if (t == NBLK - 1) *total = s[t];      // grand total
}

// ---------------------------------------------------------------------------
// Kernel 3: ordered compaction + ROI emit.
// Chunk is staged global->LDS with the CDNA5 async data mover, then ranks are
// computed with wave32 ballot prefix sums (exactly reproducing flat-index order).
// ---------------------------------------------------------------------------
__global__ void scatter_kernel(const float* __restrict__ mask,
                               const unsigned* __restrict__ offsets,
                               float* __restrict__ out) {
    __shared__ float stage[CHUNK];         // 16 KB of the WGP's 320 KB LDS
    __shared__ unsigned wsum[NWAVES];

    const int tid  = threadIdx.x;
    const int lane = tid & 31;
    const int wid  = tid >> 5;
    const long base = (long)blockIdx.x * CHUNK;
    const float* src = mask + base;        // uniform -> SGPR pair

    // --- CDNA5 async global->LDS staging (bypasses VGPRs, tracked by ASYNCcnt)
#pragma unroll
    for (int i = 0; i < ITEMS / 4; ++i) {
        const int elem = (i * THREADS + tid) * 4;          // float index in chunk
        unsigned ldsAddr = (unsigned)(uintptr_t)(&stage[elem]); // low 32 bits = LDS offset
        unsigned gOff    = (unsigned)(elem * 4);           // byte offset (GVS mode)
        asm volatile("global_load_async_to_lds_b128 %0, %1, %2"
                     :: "v"(ldsAddr), "v"(gOff), "s"(src)
                     : "memory");
    }
    asm volatile("s_wait_asynccnt 0" ::: "memory");
    __syncthreads();   // all waves' async copies complete -> LDS tile valid

    const unsigned blockBase = offsets[blockIdx.x];
    unsigned slabBase = 0;

    for (int i = 0; i < ITEMS; ++i) {
        const int li  = i * THREADS + tid;     // position inside chunk, flat order
        const float v = stage[li];
        const bool pred = v > THRESH;

        // wave32 ballot -> 32-bit mask; prefix rank within wave
        const unsigned mb     = (unsigned)__ballot(pred);
        const unsigned prefix = __popc(mb & ((1u << lane) - 1u));
        const unsigned wtot   = __popc(mb);
        if (lane == 0) wsum[wid] = wtot;
        __syncthreads();

        unsigned wbase = 0, slabTotal = 0;
#pragma unroll
        for (int w = 0; w < NWAVES; ++w) {
            unsigned s = wsum[w];
            if (w < wid) wbase += s;
            slabTotal += s;
        }

        if (pred) {
            const long idx = base + li;
            const int b   = (int)(idx >> 18);        // / (H*W)
            const int pix = (int)(idx & (HW - 1));
            const int x   = pix >> 9;                // / W
            const int y   = pix & (W_DIM - 1);       // % W
            const unsigned rank = blockBase + slabBase + wbase + prefix;
            // row: [b, x-16, y-16, b, x+16, y+16]  (24 B, 8-B aligned stride)
            float2* o = (float2*)(out + (size_t)rank * 6);
            o[0] = make_float2((float)b, (float)(x - ROI_HALF));
            o[1] = make_float2((float)(y - ROI_HALF), (float)b);
            o[2] = make_float2((float)(x + ROI_HALF), (float)(y + ROI_HALF));
        }
        slabBase += slabTotal;
        __syncthreads();   // protect wsum reuse next slab
    }
}

// ---------------------------------------------------------------------------
// Kernel 4: fill padded rows (row >= total) with -1; valid rows untouched.
// ---------------------------------------------------------------------------
__global__ void fill_kernel(const unsigned* __restrict__ total,
                            float* __restrict__ out) {
    const unsigned cnt = *total;
    const size_t r = (size_t)blockIdx.x * blockDim.x + threadIdx.x;
    if (r >= (size_t)N_ELEM || r < cnt) return;
    float2* o = (float2*)(out + r * 6);
    const float2 m1 = make_float2(-1.0f, -1.0f);
    o[0] = m1; o[1] = m1; o[2] = m1;
}

// ---------------------------------------------------------------------------
extern "C" void kernel_launch(void* const* d_in, const int* in_sizes, int n_in,
                              void* d_out, int out_size, void* d_ws, size_t ws_size,
                              hipStream_t stream) {
    const float* mask = (const float*)d_in[0];
    float* out = (float*)d_out;

    unsigned* ws       = (unsigned*)d_ws;
    unsigned* counts   = ws;                 // [NBLK]
    unsigned* offsets  = ws + NBLK;          // [NBLK]
    unsigned* total    = ws + 2 * NBLK;      // [1]

    count_kernel<<<NBLK, THREADS, 0, stream>>>(mask, counts);
    scan_kernel<<<1, NBLK, 0, stream>>>(counts, offsets, total);
    scatter_kernel<<<NBLK, THREADS, 0, stream>>>(mask, offsets, out);
    fill_kernel<<<(N_ELEM + THREADS - 1) / THREADS, THREADS, 0, stream>>>(total, out);
}